// NodeNetGNN_48369921688186
// MI455X (gfx1250) — compile-verified
//
#include <hip/hip_runtime.h>
#include <math.h>

typedef __attribute__((ext_vector_type(2))) float v2f;
typedef __attribute__((ext_vector_type(8))) float v8f;

#define N_NODE 300000
#define N_NET  60000
#define E_PIN  1200000
#define E_NEAR 2000000
#define LOG2F_ 0.6931471805599453f

// ---------------- helpers ----------------

__device__ __forceinline__ float sspf(float x) {
    // softplus(x) - log(2), numerically stable
    return fmaxf(x, 0.0f) + log1pf(expf(-fabsf(x))) - LOG2F_;
}

__device__ __forceinline__ void atomicMaxF(float* addr, float v) {
    if (v >= 0.0f) atomicMax((int*)addr, __float_as_int(v));
    else           atomicMin((unsigned int*)addr, __float_as_uint(v));
}

// 16-row tile GEMM: D[16x64] += A[16xK] @ W[Kx64], via V_WMMA_F32_16X16X4_F32.
// A row-major stride K, W row-major stride 64. acc[n] covers cols n*16..n*16+15.
// A-frag: lanes 0-15 row=q K={k0,k0+1}; lanes 16-31 row=q K={k0+2,k0+3}.
// B-frag: v0: lanes0-15 K=k0 row, lanes16-31 K=k0+2; v1: K=k0+1 / K=k0+3.
template<int K, bool FIXINF>
__device__ __forceinline__ void wmma_gemm_16xK(const float* __restrict__ A,
                                               const float* __restrict__ W,
                                               v8f acc[4], float rowscale) {
    const int lane = threadIdx.x & 31;
    const int half = lane >> 4;
    const int q    = lane & 15;
#pragma unroll
    for (int k0 = 0; k0 < K; k0 += 4) {
        const float* ap = A + q * K + k0 + 2 * half;
        v2f a;
        a.x = ap[0];
        a.y = ap[1];
        if (FIXINF) {
            a.x = (a.x < -1e37f) ? 0.0f : a.x;
            a.y = (a.y < -1e37f) ? 0.0f : a.y;
        }
        a.x *= rowscale;
        a.y *= rowscale;
#pragma unroll
        for (int n = 0; n < 4; ++n) {
            v2f b;
            b.x = W[(k0 + 2 * half + 0) * 64 + n * 16 + q];
            b.y = W[(k0 + 2 * half + 1) * 64 + n * 16 + q];
#if defined(__HIP_DEVICE_COMPILE__)
            acc[n] = __builtin_amdgcn_wmma_f32_16x16x4_f32(
                false, a, false, b, (short)0, acc[n], false, false);
#else
            (void)b;
#endif
        }
    }
}

// ---------------- generic kernels ----------------

__global__ void fill_kernel(float* __restrict__ p, int n, float v) {
    int i = blockIdx.x * 256 + threadIdx.x;
    if (i < n) p[i] = v;
}

__global__ void degree_kernel(const int* __restrict__ ps, const int* __restrict__ pd,
                              float* __restrict__ dout, float* __restrict__ din, int nE) {
    int i = blockIdx.x * 256 + threadIdx.x;
    if (i < nE) {
        atomicAdd(&dout[ps[i]], 1.0f);
        atomicAdd(&din[pd[i]], 1.0f);
    }
}

// out = epilogue(A @ W + bias). MODE: 0 = bias only, 1 = ssp, 2 = relu
template<int MODE>
__global__ void gemm_epilogue_kernel(const float* __restrict__ A, const float* __restrict__ W,
                                     const float* __restrict__ bias, float* __restrict__ out,
                                     int ntiles) {
    const int wave = threadIdx.x >> 5;
    const int tile = blockIdx.x * 8 + wave;
    if (tile >= ntiles) return;
    const int lane = threadIdx.x & 31, half = lane >> 4, q = lane & 15;
    const size_t row0 = (size_t)tile * 16;
    v8f acc[4] = {};
    wmma_gemm_16xK<64, false>(A + row0 * 64, W, acc, 1.0f);
#pragma unroll
    for (int n = 0; n < 4; ++n) {
        const float b = bias[n * 16 + q];
#pragma unroll
        for (int r = 0; r < 8; ++r) {
            float v = acc[n][r] + b;
            if (MODE == 1) v = sspf(v);
            if (MODE == 2) v = fmaxf(v, 0.0f);
            out[(row0 + r + 8 * half) * 64 + n * 16 + q] = v;
        }
    }
}

// ---------------- relation 'pins' (GraphConv) ----------------

// xw = (node_feat * rsqrt(max(deg_out,1))) @ gc_w
__global__ void gc_gemm_kernel(const float* __restrict__ nf, const float* __restrict__ deg,
                               const float* __restrict__ W, float* __restrict__ xw, int ntiles) {
    const int wave = threadIdx.x >> 5;
    const int tile = blockIdx.x * 8 + wave;
    if (tile >= ntiles) return;
    const int lane = threadIdx.x & 31, half = lane >> 4, q = lane & 15;
    const size_t row0 = (size_t)tile * 16;
    const float rs = rsqrtf(fmaxf(deg[row0 + q], 1.0f));
    v8f acc[4] = {};
    wmma_gemm_16xK<64, false>(nf + row0 * 64, W, acc, rs);
#pragma unroll
    for (int n = 0; n < 4; ++n)
#pragma unroll
        for (int r = 0; r < 8; ++r)
            xw[(row0 + r + 8 * half) * 64 + n * 16 + q] = acc[n][r];
}

__global__ void gc_scatter_kernel(const float* __restrict__ xw, const int* __restrict__ ps,
                                  const int* __restrict__ pd, float* __restrict__ net) {
    int t = blockIdx.x * 256 + threadIdx.x;  // E_PIN*64 threads exactly
    int e = t >> 6, c = t & 63;
    atomicAdd(&net[(size_t)pd[e] * 64 + c], xw[(size_t)ps[e] * 64 + c]);
}

__global__ void net_finalize_kernel(float* __restrict__ net, const float* __restrict__ din,
                                    const float* __restrict__ gcb, int n) {
    int i = blockIdx.x * 256 + threadIdx.x;
    if (i < n) {
        int row = i >> 6, c = i & 63;
        net[i] = net[i] * rsqrtf(fmaxf(din[row], 1.0f)) + gcb[c];
    }
}

// ---------------- relation 'pinned' (CFConv), fused edge MLP ----------------

__global__ void cfconv_edge_kernel(const float* __restrict__ pin_feat, const float* __restrict__ hv,
                                   const float* __restrict__ e1w, const float* __restrict__ e1b,
                                   const float* __restrict__ e2w, const float* __restrict__ e2b,
                                   const int* __restrict__ psrc, const int* __restrict__ pdst,
                                   float* __restrict__ s_out) {
    __shared__ float lds[8 * 16 * 64];
    const int wave = threadIdx.x >> 5;
    const int lane = threadIdx.x & 31, half = lane >> 4, q = lane & 15;
    const int tile = blockIdx.x * 8 + wave;   // grid sized exactly: E_PIN/16/8 blocks
    const int e0 = tile * 16;

    // t1 = ssp(pin_feat_tile @ e1w + e1b)   [16x64]
    v8f acc[4] = {};
    wmma_gemm_16xK<16, false>(pin_feat + (size_t)e0 * 16, e1w, acc, 1.0f);
    float* slab = lds + wave * 16 * 64;
#pragma unroll
    for (int n = 0; n < 4; ++n) {
        const float b = e1b[n * 16 + q];
#pragma unroll
        for (int r = 0; r < 8; ++r)
            slab[(r + 8 * half) * 64 + n * 16 + q] = sspf(acc[n][r] + b);
    }
    __syncthreads();

    // he = ssp(t1 @ e2w + e2b)
    v8f acc2[4] = {};
    wmma_gemm_16xK<64, false>(slab, e2w, acc2, 1.0f);

    int srcs[8], dsts[8];
#pragma unroll
    for (int r = 0; r < 8; ++r) {
        const int e = e0 + r + 8 * half;
        srcs[r] = psrc[e];
        dsts[r] = pdst[e];
    }
#pragma unroll
    for (int n = 0; n < 4; ++n) {
        const int col = n * 16 + q;
        const float b = e2b[col];
#pragma unroll
        for (int r = 0; r < 8; ++r) {
            const float he = sspf(acc2[n][r] + b);
            const float val = he * hv[(size_t)srcs[r] * 64 + col];
            atomicAdd(&s_out[(size_t)dsts[r] * 64 + col], val);
        }
    }
}

// ---------------- relation 'near' (SAGE pool) ----------------

__global__ void ew_kernel(const float* __restrict__ ef, const float* __restrict__ gw,
                          const float* __restrict__ gb, float* __restrict__ ew, int nE) {
    int e = blockIdx.x * 256 + threadIdx.x;
    if (e < nE) {
        const float4* p = (const float4*)(ef + (size_t)e * 16);
        const float4* w = (const float4*)gw;
        float s = 0.0f;
#pragma unroll
        for (int i = 0; i < 4; ++i) {
            float4 a = p[i], b = w[i];
            s += a.x * b.x + a.y * b.y + a.z * b.z + a.w * b.w;
        }
        ew[e] = 1.0f / (1.0f + expf(-(s + gb[0])));
    }
}

__global__ void sage_scatter_kernel(const float* __restrict__ hp, const float* __restrict__ ew,
                                    const int* __restrict__ ns, const int* __restrict__ nd,
                                    float* __restrict__ mx) {
    int t = blockIdx.x * 256 + threadIdx.x;  // E_NEAR*64 threads exactly
    int e = t >> 6, c = t & 63;
    float v = hp[(size_t)ns[e] * 64 + c] * ew[e];
    atomicMaxF(&mx[(size_t)nd[e] * 64 + c], v);
}

// node_out = max(cf_out(in d_out), node_feat@self_w + self_b + fixinf(mx)@neigh_w)
__global__ void final_fused_kernel(const float* __restrict__ nf, const float* __restrict__ mx,
                                   const float* __restrict__ self_w, const float* __restrict__ self_b,
                                   const float* __restrict__ neigh_w, float* __restrict__ out,
                                   int ntiles) {
    const int wave = threadIdx.x >> 5;
    const int tile = blockIdx.x * 8 + wave;
    if (tile >= ntiles) return;
    const int lane = threadIdx.x & 31, half = lane >> 4, q = lane & 15;
    const size_t row0 = (size_t)tile * 16;
    v8f acc[4] = {};
    wmma_gemm_16xK<64, false>(nf + row0 * 64, self_w, acc, 1.0f);
    wmma_gemm_16xK<64, true>(mx + row0 * 64, neigh_w, acc, 1.0f);
#pragma unroll
    for (int n = 0; n < 4; ++n) {
        const float b = self_b[n * 16 + q];
#pragma unroll
        for (int r = 0; r < 8; ++r) {
            const size_t idx = (row0 + r + 8 * half) * 64 + n * 16 + q;
            out[idx] = fmaxf(out[idx], acc[n][r] + b);
        }
    }
}

// ---------------- launch ----------------

extern "C" void kernel_launch(void* const* d_in, const int* in_sizes, int n_in,
                              void* d_out, int out_size, void* d_ws, size_t ws_size,
                              hipStream_t stream) {
    const float* node_feat   = (const float*)d_in[0];
    const float* net_feat    = (const float*)d_in[1];
    const float* pin_feat    = (const float*)d_in[2];
    const float* edge_feat   = (const float*)d_in[3];
    const float* gc_w        = (const float*)d_in[4];
    const float* gc_b        = (const float*)d_in[5];
    const float* cf_node_w   = (const float*)d_in[6];
    const float* cf_node_b   = (const float*)d_in[7];
    const float* cf_e1_w     = (const float*)d_in[8];
    const float* cf_e1_b     = (const float*)d_in[9];
    const float* cf_e2_w     = (const float*)d_in[10];
    const float* cf_e2_b     = (const float*)d_in[11];
    const float* cf_out_w    = (const float*)d_in[12];
    const float* cf_out_b    = (const float*)d_in[13];
    const float* sage_pool_w = (const float*)d_in[14];
    const float* sage_pool_b = (const float*)d_in[15];
    const float* sage_self_w = (const float*)d_in[16];
    const float* sage_self_b = (const float*)d_in[17];
    const float* sage_neigh_w= (const float*)d_in[18];
    const float* geom_w      = (const float*)d_in[19];
    const float* geom_b      = (const float*)d_in[20];
    const int* pins_src      = (const int*)d_in[21];
    const int* pins_dst      = (const int*)d_in[22];
    const int* pinned_src    = (const int*)d_in[23];
    const int* pinned_dst    = (const int*)d_in[24];
    const int* near_src      = (const int*)d_in[25];
    const int* near_dst      = (const int*)d_in[26];

    float* out      = (float*)d_out;
    float* node_out = out;                          // [N_NODE*64]
    float* net_out  = out + (size_t)N_NODE * 64;    // [N_NET*64]

    // workspace layout (floats)
    float* ws      = (float*)d_ws;
    float* deg_out = ws;                                   // N_NODE
    float* deg_in  = deg_out + N_NODE;                     // N_NET
    float* bufA    = deg_in + N_NET;                       // N_NODE*64  (xw, later h_pool)
    float* bufB    = bufA + (size_t)N_NODE * 64;           // N_NODE*64  (s, later mx)
    float* hv      = bufB + (size_t)N_NODE * 64;           // N_NET*64
    float* ew      = hv + (size_t)N_NET * 64;              // E_NEAR

    const int NODE_TILES = N_NODE / 16;   // 18750
    const int NET_TILES  = N_NET / 16;    // 3750
    const int PIN_TILES  = E_PIN / 16;    // 75000

    // ---- init ----
    fill_kernel<<<(N_NODE + 255) / 256, 256, 0, stream>>>(deg_out, N_NODE, 0.0f);
    fill_kernel<<<(N_NET + 255) / 256, 256, 0, stream>>>(deg_in, N_NET, 0.0f);
    fill_kernel<<<(N_NET * 64 + 255) / 256, 256, 0, stream>>>(net_out, N_NET * 64, 0.0f);
    fill_kernel<<<(N_NODE * 64 + 255) / 256, 256, 0, stream>>>(bufB, N_NODE * 64, 0.0f);

    // ---- 'pins': GraphConv ----
    degree_kernel<<<(E_PIN + 255) / 256, 256, 0, stream>>>(pins_src, pins_dst, deg_out, deg_in, E_PIN);
    gc_gemm_kernel<<<(NODE_TILES + 7) / 8, 256, 0, stream>>>(node_feat, deg_out, gc_w, bufA, NODE_TILES);
    gc_scatter_kernel<<<E_PIN * 64 / 256, 256, 0, stream>>>(bufA, pins_src, pins_dst, net_out);
    net_finalize_kernel<<<(N_NET * 64 + 255) / 256, 256, 0, stream>>>(net_out, deg_in, gc_b, N_NET * 64);

    // ---- 'pinned': CFConv ----
    gemm_epilogue_kernel<0><<<(NET_TILES + 7) / 8, 256, 0, stream>>>(net_feat, cf_node_w, cf_node_b, hv, NET_TILES);
    cfconv_edge_kernel<<<PIN_TILES / 8, 256, 0, stream>>>(pin_feat, hv, cf_e1_w, cf_e1_b,
                                                          cf_e2_w, cf_e2_b, pinned_src, pinned_dst, bufB);
    gemm_epilogue_kernel<1><<<(NODE_TILES + 7) / 8, 256, 0, stream>>>(bufB, cf_out_w, cf_out_b, node_out, NODE_TILES);

    // ---- 'near': SAGE pool ----
    fill_kernel<<<(N_NODE * 64 + 255) / 256, 256, 0, stream>>>(bufB, N_NODE * 64, -INFINITY);
    ew_kernel<<<(E_NEAR + 255) / 256, 256, 0, stream>>>(edge_feat, geom_w, geom_b, ew, E_NEAR);
    gemm_epilogue_kernel<2><<<(NODE_TILES + 7) / 8, 256, 0, stream>>>(node_feat, sage_pool_w, sage_pool_b, bufA, NODE_TILES);
    sage_scatter_kernel<<<E_NEAR * 64 / 256, 256, 0, stream>>>(bufA, ew, near_src, near_dst, bufB);

    // ---- final: node_out = max(cf_out, sage_out) ----
    final_fused_kernel<<<(NODE_TILES + 7) / 8, 256, 0, stream>>>(node_feat, bufB, sage_self_w,
                                                                 sage_self_b, sage_neigh_w,
                                                                 node_out, NODE_TILES);
}